// Gen_15144054686380
// MI455X (gfx1250) — compile-verified
//
#include <hip/hip_runtime.h>

typedef __attribute__((ext_vector_type(2))) float v2f;
typedef __attribute__((ext_vector_type(8))) float v8f;

#define B_     16
#define QL_    16
#define H_     32
#define D_     128
#define CACHE_ 4096
#define LFULL_ 4112
#define NBLK_  257     // 4112 / 16 key blocks; block 0 = fresh k/v
#define NWAVE_ 4

// Softmax update + P transpose + O += P*V for one 16-key block.
// VS = row stride (in floats) of the V source block; compile-time constant so
// all fragment loads become immediate-offset vector loads.
template <int VS>
__device__ __forceinline__ void attn_tail(v8f s, const float* __restrict__ vbase,
                                          float* __restrict__ pws, int hf, int ln,
                                          float* __restrict__ m_run,
                                          float* __restrict__ l_run,
                                          v8f* __restrict__ o)
{
  const float inv_d = 1.0f / 128.0f;
  float p[8], sc[8];
  #pragma unroll
  for (int r = 0; r < 8; ++r) {
    const float sv = s[r] * inv_d;
    float bm = sv;
    bm = fmaxf(bm, __shfl_xor(bm, 1));
    bm = fmaxf(bm, __shfl_xor(bm, 2));
    bm = fmaxf(bm, __shfl_xor(bm, 4));
    bm = fmaxf(bm, __shfl_xor(bm, 8));
    const float mn = fmaxf(m_run[r], bm);
    sc[r] = __expf(m_run[r] - mn);
    const float pv = __expf(sv - mn);
    float rs = pv;
    rs += __shfl_xor(rs, 1);
    rs += __shfl_xor(rs, 2);
    rs += __shfl_xor(rs, 4);
    rs += __shfl_xor(rs, 8);
    l_run[r] = l_run[r] * sc[r] + rs;
    m_run[r] = mn;
    p[r] = pv;
  }
  #pragma unroll
  for (int c = 0; c < 8; ++c)
    #pragma unroll
    for (int r = 0; r < 8; ++r)
      o[c][r] *= sc[r];

  // P: C-layout -> A-layout via per-wave LDS tile (wave-local, DS is in-order)
  #pragma unroll
  for (int r = 0; r < 8; ++r)
    pws[(r + 8 * hf) * 16 + ln] = p[r];
  asm volatile("s_wait_dscnt 0x0" ::: "memory");
  v2f pa[4];
  #pragma unroll
  for (int s4 = 0; s4 < 4; ++s4) {
    const int kx = 4 * s4 + 2 * hf;
    pa[s4] = *(const v2f*)(pws + ln * 16 + kx);   // ds_load_b64
  }

  // O += P * V_blk : 8 d-chunks x 4 k-steps, constant-offset B fragments
  #pragma unroll
  for (int c = 0; c < 8; ++c) {
    #pragma unroll
    for (int s4 = 0; s4 < 4; ++s4) {
      const int kr = 4 * s4 + 2 * hf;
      v2f bf;
      bf.x = vbase[kr * VS + 16 * c + ln];
      bf.y = vbase[(kr + 1) * VS + 16 * c + ln];
      o[c] = __builtin_amdgcn_wmma_f32_16x16x4_f32(false, pa[s4], false, bf,
                                                   (short)0, o[c], false, false);
    }
  }
}

__global__ __launch_bounds__(128, 1) void attn_fused(
    const float* __restrict__ q, const float* __restrict__ k,
    const float* __restrict__ v, const float* __restrict__ cacheK,
    const float* __restrict__ cacheV,
    float* __restrict__ res, float* __restrict__ kfull, float* __restrict__ vfull)
{
  __shared__ float qlds[16 * 128];       // Q tile, shared by all 4 waves
  __shared__ float p_ws[NWAVE_ * 256];   // per-wave 16x16 P transpose tile
  __shared__ float o_acc[16 * 128];      // cross-wave output accumulator
  __shared__ float mbuf[NWAVE_ * 16];
  __shared__ float lbuf[NWAVE_ * 16];
  __shared__ float lgbuf[16];

  const int bh   = blockIdx.x;
  const int b    = bh / H_;
  const int h    = bh % H_;
  const int tid  = threadIdx.x;
  const int lane = tid & 31;
  const int wave = tid >> 5;
  const int hf   = lane >> 4;
  const int ln   = lane & 15;

  // cooperative init: zero O accumulator, stage Q tile into LDS
  for (int i = tid; i < 16 * 128; i += 128) o_acc[i] = 0.0f;
  for (int i = tid; i < 512; i += 128) {
    const int m = i >> 5, c4 = i & 31;
    ((float4*)qlds)[i] =
        ((const float4*)(q + ((size_t)(b * QL_ + m) * H_ + h) * D_))[c4];
  }
  __syncthreads();

  v8f o[8];
  #pragma unroll
  for (int c = 0; c < 8; ++c) o[c] = (v8f){};
  float m_run[8], l_run[8];
  #pragma unroll
  for (int r = 0; r < 8; ++r) { m_run[r] = -__builtin_inff(); l_run[r] = 0.0f; }

  const size_t kc_base = (size_t)bh * D_ * CACHE_;
  const size_t kf_base = (size_t)bh * D_ * LFULL_;
  const size_t vc_base = (size_t)bh * CACHE_ * D_;
  const size_t vf_base = (size_t)bh * LFULL_ * D_;
  float* pws = p_ws + wave * 256;
  const float* qrow = qlds + ln * 128;

  // ---------------- block 0: fresh k/v (wave 0 only; wave-uniform) ----------
  if (wave == 0) {
    for (int i = lane; i < 2048; i += 32) {            // new K, transposed scatter
      const int d = i >> 4, j = i & 15;
      kfull[kf_base + (size_t)d * LFULL_ + j] =
          k[((size_t)(b * QL_ + j) * H_ + h) * D_ + d];
    }
    for (int i = lane; i < 512; i += 32) {             // new V rows, float4
      const int r = i >> 5, c4 = i & 31;
      const float4 val =
          ((const float4*)(v + ((size_t)(b * QL_ + r) * H_ + h) * D_))[c4];
      ((float4*)(vfull + vf_base + (size_t)r * D_))[c4] = val;
    }
    // S = Q * Knew : K column ln is contiguous in d -> v2f loads
    const float* kb = k + ((size_t)(b * QL_ + ln) * H_ + h) * D_;
    v8f s = (v8f){};
    #pragma unroll
    for (int kk = 0; kk < 32; ++kk) {
      const int d0 = 4 * kk + 2 * hf;
      const v2f pa = *(const v2f*)(qrow + d0);         // ds_load_b64
      const v2f bf = *(const v2f*)(kb + d0);           // global_load_b64
      s = __builtin_amdgcn_wmma_f32_16x16x4_f32(false, pa, false, bf,
                                                (short)0, s, false, false);
    }
    attn_tail<H_ * D_>(s, v + ((size_t)(b * QL_) * H_ + h) * D_,
                       pws, hf, ln, m_run, l_run, o);
  }

  // ---------------- cache blocks 1..256, constant strides -------------------
  for (int blk = 1 + wave; blk < NBLK_; blk += NWAVE_) {
    const int n0 = blk * 16;
    const int j0 = n0 - 16;                            // column offset in cache

    // prefetch this wave's next block (global_prefetch_b8)
    if (blk + NWAVE_ < NBLK_) {
      const int jn = j0 + NWAVE_ * 16;
      __builtin_prefetch(cacheV + vc_base + (size_t)jn * D_ + lane * 32, 0, 1);
      __builtin_prefetch(cacheV + vc_base + (size_t)jn * D_ + 1024 + lane * 32, 0, 1);
      #pragma unroll
      for (int pr = 0; pr < 4; ++pr)
        __builtin_prefetch(cacheK + kc_base + (size_t)(pr * 32 + lane) * CACHE_ + jn, 0, 1);
    }

    // fused cache-append copy (coalesced float4), leaves block hot in cache
    for (int i = lane; i < 512; i += 32) {             // K: 128 d-rows x 16 cols
      const int d = i >> 2, c4 = i & 3;
      const float4 val =
          ((const float4*)(cacheK + kc_base + (size_t)d * CACHE_ + j0))[c4];
      ((float4*)(kfull + kf_base + (size_t)d * LFULL_ + n0))[c4] = val;
    }
    for (int i = lane; i < 512; i += 32) {             // V: 16 rows x 128
      const int r = i >> 5, c4 = i & 31;
      const float4 val =
          ((const float4*)(cacheV + vc_base + (size_t)(j0 + r) * D_))[c4];
      ((float4*)(vfull + vf_base + (size_t)(n0 + r) * D_))[c4] = val;
    }

    // S = Q * K_blk : one base register + immediate offsets (d0 * CACHE_)
    const float* kb = cacheK + kc_base + j0 + ln;
    v8f s = (v8f){};
    #pragma unroll
    for (int kk = 0; kk < 32; ++kk) {
      const int d0 = 4 * kk + 2 * hf;
      const v2f pa = *(const v2f*)(qrow + d0);         // ds_load_b64
      v2f bf;
      bf.x = kb[d0 * CACHE_];
      bf.y = kb[(d0 + 1) * CACHE_];
      s = __builtin_amdgcn_wmma_f32_16x16x4_f32(false, pa, false, bf,
                                                (short)0, s, false, false);
    }
    attn_tail<D_>(s, cacheV + vc_base + (size_t)j0 * D_,
                  pws, hf, ln, m_run, l_run, o);
  }

  // ---------------- cross-wave flash merge ----------------------------------
  #pragma unroll
  for (int r = 0; r < 8; ++r) {
    mbuf[wave * 16 + r + 8 * hf] = m_run[r];
    lbuf[wave * 16 + r + 8 * hf] = l_run[r];
  }
  __syncthreads();

  float gsc[8];
  #pragma unroll
  for (int r = 0; r < 8; ++r) {
    const int row = r + 8 * hf;
    float Mg = -__builtin_inff();
    for (int w = 0; w < NWAVE_; ++w) Mg = fmaxf(Mg, mbuf[w * 16 + row]);
    float Lg = 0.0f;
    for (int w = 0; w < NWAVE_; ++w)
      Lg += lbuf[w * 16 + row] * __expf(mbuf[w * 16 + row] - Mg);
    gsc[r] = __expf(m_run[r] - Mg);
    lgbuf[row] = Lg;   // identical value from every wave
  }

  #pragma unroll
  for (int c = 0; c < 8; ++c)
    #pragma unroll
    for (int r = 0; r < 8; ++r)
      atomicAdd(&o_acc[(r + 8 * hf) * 128 + 16 * c + ln], o[c][r] * gsc[r]);

  __syncthreads();

  for (int i = tid; i < 2048; i += 128) {
    const int row = i >> 7, d = i & 127;
    res[((size_t)(b * QL_ + row) * H_ + h) * D_ + d] = o_acc[i] / lgbuf[row];
  }
}

extern "C" void kernel_launch(void* const* d_in, const int* in_sizes, int n_in,
                              void* d_out, int out_size, void* d_ws, size_t ws_size,
                              hipStream_t stream) {
  const float* q  = (const float*)d_in[0];
  const float* k  = (const float*)d_in[1];
  const float* v  = (const float*)d_in[2];
  const float* cK = (const float*)d_in[3];
  const float* cV = (const float*)d_in[4];

  float* res   = (float*)d_out;
  float* kfull = res + (size_t)B_ * QL_ * H_ * D_;
  float* vfull = kfull + (size_t)B_ * H_ * D_ * LFULL_;

  attn_fused<<<B_ * H_, 128, 0, stream>>>(q, k, v, cK, cV, res, kfull, vfull);
}